// HistoMaker_25950192403260
// MI455X (gfx1250) — compile-verified
//
#include <hip/hip_runtime.h>
#include <hip/hip_bf16.h>

typedef float v2f __attribute__((ext_vector_type(2)));
typedef float v8f __attribute__((ext_vector_type(8)));

// sech(x) = 2u / (1 + u^2), u = exp(-|x|) in (0,1]; denominator in [1,2] so a
// raw v_rcp_f32 (~1 ulp) is safe — avoids the IEEE div_scale/div_fixup chain.
// sech(+-inf)=0 also implements the reference's isinf->0 mask (taps nonzero).
__device__ __forceinline__ float sech_fast(float x) {
    float u = __expf(-fabsf(x));
    float r = __builtin_amdgcn_rcpf(__builtin_fmaf(u, u, 1.0f));
    return 2.0f * u * r;
}

// Each wave owns 16-pixel tiles. One V_WMMA_F32_16X16X4_F32 per input channel
// computes X[m,o] = y[m]*k[o] + b[o] exactly in f32; sech + channel-sum in VALU.
__global__ void __launch_bounds__(256) histo_wmma_kernel(
    const float* __restrict__ img, const float* __restrict__ kern,
    const float* __restrict__ bias, float* __restrict__ out, long tiles)
{
    const int  lane  = threadIdx.x & 31;
    const int  wave  = threadIdx.x >> 5;
    const long gwave = (long)blockIdx.x * (blockDim.x >> 5) + wave;
    const long nwave = (long)gridDim.x * (blockDim.x >> 5);
    const int  n     = lane & 15;      // output column this lane owns in C/D
    const bool lo    = lane < 16;

    // B matrix (4x16): row K=0 = conv kernel taps, rows K=1..3 zero.
    // VGPR0: lanes0-15 K=0, lanes16-31 K=2 (zero); VGPR1: K=1/K=3 (zero).
    v2f bm;
    bm.x = (lane < 11) ? kern[lane] : 0.0f;
    bm.y = 0.0f;

    // C matrix (16x16): bias[n] broadcast down M.
    float bn = (n < 11) ? bias[n] : 0.0f;
    v8f cm;
#pragma unroll
    for (int j = 0; j < 8; ++j) cm[j] = bn;

    for (long t = gwave; t < tiles; t += nwave) {
        const long  pix0 = t * 16;
        const float* src = img + (pix0 + (long)(lane & 15)) * 3;
        // prefetch this wave's next tile into near cache (global_prefetch_b8)
        __builtin_prefetch(src + nwave * 48, 0, 3);

        float y0 = src[0];
        float y1 = src[1];
        float y2 = src[2];

        v8f acc;
#pragma unroll
        for (int j = 0; j < 8; ++j) acc[j] = 0.0f;

#pragma unroll
        for (int c = 0; c < 3; ++c) {
            float y = (c == 0) ? y0 : (c == 1) ? y1 : y2;
            // A (16x4): lanes0-15 VGPR0 = K=0 column = y; everything else 0.
            v2f am;
            am.x = lo ? y : 0.0f;
            am.y = 0.0f;
            v8f d = __builtin_amdgcn_wmma_f32_16x16x4_f32(
                false, am, false, bm, (short)0, cm, false, false);
#pragma unroll
            for (int j = 0; j < 8; ++j) acc[j] += sech_fast(d[j]);
        }

        // D layout: VGPR j -> (M = j or j+8, N = lane&15). Store N<11 only.
        if (n < 11) {
            const int  mbase = lo ? 0 : 8;
            float*     op    = out + (pix0 + mbase) * 11 + n;
#pragma unroll
            for (int j = 0; j < 8; ++j) op[(long)j * 11] = acc[j];
        }
    }
}

// Scalar fallback for a (pixel % 16) tail — unused for the 16x512x512 shape
// but keeps kernel_launch correct for any size.
__global__ void histo_tail_kernel(
    const float* __restrict__ img, const float* __restrict__ kern,
    const float* __restrict__ bias, float* __restrict__ out,
    long start_pix, long n_pix)
{
    long p = start_pix + (long)blockIdx.x * blockDim.x + threadIdx.x;
    if (p >= n_pix) return;
    float y0 = img[p * 3 + 0];
    float y1 = img[p * 3 + 1];
    float y2 = img[p * 3 + 2];
#pragma unroll
    for (int o = 0; o < 11; ++o) {
        float k = kern[o], b = bias[o];
        float s = sech_fast(fmaf(y0, k, b)) +
                  sech_fast(fmaf(y1, k, b)) +
                  sech_fast(fmaf(y2, k, b));
        out[p * 11 + o] = s;
    }
}

extern "C" void kernel_launch(void* const* d_in, const int* in_sizes, int n_in,
                              void* d_out, int out_size, void* d_ws, size_t ws_size,
                              hipStream_t stream) {
    (void)n_in; (void)out_size; (void)d_ws; (void)ws_size;
    const float* img  = (const float*)d_in[0];
    const float* kern = (const float*)d_in[1];
    const float* bias = (const float*)d_in[2];
    float*       out  = (float*)d_out;

    const long n_pix = (long)in_sizes[0] / 3;   // 4,194,304 for 16x512x512x3
    const long tiles = n_pix / 16;              // full 16-pixel WMMA tiles

    if (tiles > 0) {
        const int  threads       = 256;         // 8 wave32 per block
        const int  wavesPerBlock = threads / 32;
        // ~8 tiles per wave
        long waves  = (tiles + 7) / 8;
        long blocks = (waves + wavesPerBlock - 1) / wavesPerBlock;
        if (blocks < 1)      blocks = 1;
        if (blocks > 65535L) blocks = 65535L;
        histo_wmma_kernel<<<(int)blocks, threads, 0, stream>>>(
            img, kern, bias, out, tiles);
    }

    const long rem = n_pix - tiles * 16;
    if (rem > 0) {
        int rblocks = (int)((rem + 255) / 256);
        histo_tail_kernel<<<rblocks, 256, 0, stream>>>(
            img, kern, bias, out, tiles * 16, n_pix);
    }
}